// Blur_32701880991938
// MI455X (gfx1250) — compile-verified
//
#include <hip/hip_runtime.h>
#include <hip/hip_bf16.h>
#include <stdint.h>

// Upsample-by-2 + [1,3,3,1]/8 separable blur (upfirdn2d), NCHW depthwise.
// Input  x  : (16, 128, 128, 128) f32
// Output out: (16, 128, 256, 256) f32
//
// Memory-bound: ~0.67 GB moved @ 23.3 TB/s => ~29 us floor. Compute (~1 GFLOP
// after polyphase factorization) is negligible, so no WMMA; the CDNA5 path
// that matters is async global->LDS tile staging + wide coalesced stores.

#define TW 64            // input tile width per block
#define TH 8             // input tile height per block
#define LW (TW + 2)      // 66 (halo of 1 each side)
#define LH (TH + 2)      // 10
#define LDS_ELEMS (LW * LH)  // 660 floats = 2640 B

__global__ __launch_bounds__(256) void blur_up2_kernel(
    const float* __restrict__ x, float* __restrict__ out, int H, int W) {
  __shared__ float tile[LDS_ELEMS];

  const int tid   = threadIdx.x;
  const int ib_x  = blockIdx.x * TW;   // input col base of tile
  const int ib_y  = blockIdx.y * TH;   // input row base of tile
  const int plane = blockIdx.z;        // n*c plane

  const float* xp = x + (size_t)plane * H * W;

  // ---- Stage input tile (+halo) into LDS via CDNA5 async global->LDS ----
  for (int i = tid; i < LDS_ELEMS; i += 256) {
    const int r  = i / LW;
    const int c  = i - r * LW;
    const int gy = ib_y + r - 1;
    const int gx = ib_x + c - 1;
    // Generic LDS pointer: low 32 bits are the LDS byte offset (aperture
    // scheme), which is what the async-load VDST operand wants.
    const unsigned lds_byte = (unsigned)(uintptr_t)(&tile[i]);
    if ((unsigned)gy < (unsigned)H && (unsigned)gx < (unsigned)W) {
      const unsigned long long gaddr =
          (unsigned long long)(uintptr_t)(xp + (size_t)gy * W + gx);
      // GLOBAL_LOAD_ASYNC_TO_LDS_B32 (tracked by ASYNCcnt), per-lane, EXEC-
      // predicated by the surrounding branch.
      asm volatile("global_load_async_to_lds_b32 %0, %1, off"
                   :
                   : "v"(lds_byte), "v"(gaddr)
                   : "memory");
    } else {
      tile[i] = 0.0f;  // zero-pad halo outside the image
    }
  }
  asm volatile("s_wait_asynccnt 0" ::: "memory");
  __syncthreads();

  // ---- Polyphase compute: each thread -> 4x2 output block ----
  // tx covers input cols (2tx, 2tx+1); ty covers input row ty.
  const int tx   = tid & 31;
  const int ty   = tid >> 5;
  const int base = ty * LW + 2 * tx;  // LDS row ty == input row (ib_y+ty-1),
                                      // LDS col 2tx == input col (ib_x+2tx-1)
  float4 h[3];
#pragma unroll
  for (int r = 0; r < 3; ++r) {
    const float a = tile[base + r * LW + 0];
    const float b = tile[base + r * LW + 1];
    const float c = tile[base + r * LW + 2];
    const float d = tile[base + r * LW + 3];
    // Horizontal 2-tap polyphase (un-normalized weights 1,3):
    h[r].x = a + 3.0f * b;        // even col 4tx
    h[r].y = 3.0f * b + c;        // odd  col 4tx+1
    h[r].z = b + 3.0f * c;        // even col 4tx+2
    h[r].w = 3.0f * c + d;        // odd  col 4tx+3
  }
  const float s = 1.0f / 64.0f;   // (1/8)*(1/8) normalization
  float4 o0, o1;
  o0.x = (h[0].x + 3.0f * h[1].x) * s;
  o0.y = (h[0].y + 3.0f * h[1].y) * s;
  o0.z = (h[0].z + 3.0f * h[1].z) * s;
  o0.w = (h[0].w + 3.0f * h[1].w) * s;
  o1.x = (3.0f * h[1].x + h[2].x) * s;
  o1.y = (3.0f * h[1].y + h[2].y) * s;
  o1.z = (3.0f * h[1].z + h[2].z) * s;
  o1.w = (3.0f * h[1].w + h[2].w) * s;

  // ---- Coalesced 128-bit stores ----
  const int OW = W * 2;
  const int OH = H * 2;
  float* op    = out + (size_t)plane * OH * OW;
  const int oy = (ib_y + ty) * 2;
  const int ox = (ib_x + 2 * tx) * 2;   // multiple of 4 -> 16B aligned
  *(float4*)(op + (size_t)oy * OW + ox)       = o0;
  *(float4*)(op + (size_t)(oy + 1) * OW + ox) = o1;
}

extern "C" void kernel_launch(void* const* d_in, const int* in_sizes, int n_in,
                              void* d_out, int out_size, void* d_ws,
                              size_t ws_size, hipStream_t stream) {
  (void)in_sizes; (void)n_in; (void)out_size; (void)d_ws; (void)ws_size;
  const float* x = (const float*)d_in[0];
  float* out     = (float*)d_out;

  const int N = 16, C = 128, H = 128, W = 128;
  dim3 grid(W / TW, H / TH, N * C);  // (2, 16, 2048)
  blur_up2_kernel<<<grid, dim3(256), 0, stream>>>(x, out, H, W);
}